// _TripletMessageLight_16784732193363
// MI455X (gfx1250) — compile-verified
//
#include <hip/hip_runtime.h>
#include <hip/hip_bf16.h>
#include <math.h>

typedef __attribute__((ext_vector_type(2))) float v2f;
typedef __attribute__((ext_vector_type(8))) float v8f;

#define C_DIM 64
#define CE_DIM 16
#define NEG_SLOPE 0.2f

// ---------------------------------------------------------------------------
// Branch-safe float atomic max: signed-int max for non-negative values,
// unsigned-int min for negative values. Correct for IEEE-754 with -inf init.
// ---------------------------------------------------------------------------
__device__ __forceinline__ void atomicMaxFloat(float* addr, float v) {
    if (v >= 0.0f) {
        atomicMax((int*)addr, __float_as_int(v));
    } else {
        atomicMin((unsigned int*)addr, __float_as_uint(v));
    }
}

// ---------------------------------------------------------------------------
// Init: out[n,c] = bias[c]; m[n] = -inf; ssum[n] = 0
// ---------------------------------------------------------------------------
__global__ void tml_init_kernel(float* __restrict__ out, const float* __restrict__ bias,
                                float* __restrict__ m, float* __restrict__ ssum, int N) {
    int i = blockIdx.x * blockDim.x + threadIdx.x;
    if (i < N * C_DIM) out[i] = bias[i & (C_DIM - 1)];
    if (i < N) {
        m[i] = -INFINITY;
        ssum[i] = 0.0f;
    }
}

// ---------------------------------------------------------------------------
// h = x @ W via V_WMMA_F32_16X16X4_F32 (wave32, one 16x16 tile per wave).
// A layout (16x4 f32): lanes 0-15 -> M=lane, K={k0,k0+1}; lanes 16-31 -> K={k0+2,k0+3}
// B layout (4x16 f32): VGPR v -> row K=k0+v (lanes 0-15) / K=k0+v+2 (lanes 16-31), col=lane%16
// C layout (16x16 f32): VGPR r -> row r (lanes 0-15) / row r+8 (lanes 16-31), col=lane%16
// ---------------------------------------------------------------------------
__global__ void tml_gemm_kernel(const float* __restrict__ x, const float* __restrict__ W,
                                float* __restrict__ h, int N) {
    const int tilesN = C_DIM / 16; // 4
    int waveId = blockIdx.x * (blockDim.x / 32) + (threadIdx.x >> 5);
    int lane   = threadIdx.x & 31;
    int tileM  = waveId / tilesN;
    int tileN  = waveId % tilesN;
    if (tileM * 16 >= N) return;   // wave-uniform: EXEC stays all-ones for WMMA

    int half   = lane >> 4;        // 0 or 1
    int lane16 = lane & 15;
    int rowA   = tileM * 16 + lane16;
    int colB   = tileN * 16 + lane16;

    v8f acc = {};
    #pragma unroll
    for (int k0 = 0; k0 < C_DIM; k0 += 4) {
        v2f a, b;
        int ka = k0 + 2 * half;
        a[0] = x[rowA * C_DIM + ka];
        a[1] = x[rowA * C_DIM + ka + 1];
        b[0] = W[(ka + 0) * C_DIM + colB];
        b[1] = W[(ka + 1) * C_DIM + colB];
        acc = __builtin_amdgcn_wmma_f32_16x16x4_f32(
            /*neg_a=*/false, a, /*neg_b=*/false, b,
            /*c_mod=*/(short)0, acc, /*reuse_a=*/false, /*reuse_b=*/false);
    }

    int rowBase = tileM * 16 + 8 * half;
    #pragma unroll
    for (int r = 0; r < 8; ++r) {
        h[(rowBase + r) * C_DIM + colB] = acc[r];
    }
}

// ---------------------------------------------------------------------------
// Per-node attention scalars: s1[n] = h[n] . wa[0:64], s3[n] = h[n] . wa[80:144]
// (factors the per-edge 144-wide dot into cheap per-node precompute)
// ---------------------------------------------------------------------------
__global__ void tml_node_scalars_kernel(const float* __restrict__ h,
                                        const float* __restrict__ w_att,
                                        float* __restrict__ s1, float* __restrict__ s3, int N) {
    int n = blockIdx.x * blockDim.x + threadIdx.x;
    if (n >= N) return;
    const float* hp = h + (size_t)n * C_DIM;
    float a1 = 0.0f, a3 = 0.0f;
    #pragma unroll
    for (int c = 0; c < C_DIM; ++c) {
        float hv = hp[c];
        a1 = fmaf(hv, w_att[c], a1);
        a3 = fmaf(hv, w_att[C_DIM + CE_DIM + c], a3);
    }
    s1[n] = a1;
    s3[n] = a3;
}

// ---------------------------------------------------------------------------
// alpha[e] = leaky_relu(s1[dst] + eattr[e].wa_mid + s3[src]); atomic segment max
// ---------------------------------------------------------------------------
__global__ void tml_alpha_kernel(const int* __restrict__ ei, const float* __restrict__ eattr,
                                 const float* __restrict__ w_att,
                                 const float* __restrict__ s1, const float* __restrict__ s3,
                                 float* __restrict__ alpha, float* __restrict__ m, int E) {
    int e = blockIdx.x * blockDim.x + threadIdx.x;
    if (e >= E) return;
    int src = ei[e];
    int dst = ei[E + e];
    float a = s1[dst] + s3[src];
    const float* ep = eattr + (size_t)e * CE_DIM;
    #pragma unroll
    for (int c = 0; c < CE_DIM; ++c) {
        a = fmaf(ep[c], w_att[C_DIM + c], a);
    }
    a = (a > 0.0f) ? a : NEG_SLOPE * a;
    alpha[e] = a;
    atomicMaxFloat(m + dst, a);
}

// ---------------------------------------------------------------------------
// a[e] = exp(alpha[e] - m[dst]); atomic segment sum (in-place on alpha buffer)
// ---------------------------------------------------------------------------
__global__ void tml_exp_kernel(const int* __restrict__ ei, float* __restrict__ alpha,
                               const float* __restrict__ m, float* __restrict__ ssum, int E) {
    int e = blockIdx.x * blockDim.x + threadIdx.x;
    if (e >= E) return;
    int dst = ei[E + e];
    float a = __expf(alpha[e] - m[dst]);
    alpha[e] = a;
    atomicAdd(ssum + dst, a);
}

// ---------------------------------------------------------------------------
// out[dst] += (a[e] / (ssum[dst] + 1e-16)) * h[src]; one thread per (edge, ch),
// so each wave covers half of one edge row (32 of 64 channels) -> coalesced gathers and coalesced atomics.
// ---------------------------------------------------------------------------
__global__ void tml_scatter_kernel(const int* __restrict__ ei, const float* __restrict__ a,
                                   const float* __restrict__ ssum, const float* __restrict__ h,
                                   float* __restrict__ out, int E) {
    long long idx = (long long)blockIdx.x * blockDim.x + threadIdx.x;
    if (idx >= (long long)E * C_DIM) return;
    int e = (int)(idx >> 6);
    int c = (int)(idx & (C_DIM - 1));
    int src = ei[e];
    int dst = ei[E + e];
    float coef = a[e] / (ssum[dst] + 1e-16f);
    atomicAdd(out + (size_t)dst * C_DIM + c, coef * h[(size_t)src * C_DIM + c]);
}

// ---------------------------------------------------------------------------
extern "C" void kernel_launch(void* const* d_in, const int* in_sizes, int n_in,
                              void* d_out, int out_size, void* d_ws, size_t ws_size,
                              hipStream_t stream) {
    const float* x     = (const float*)d_in[0];
    const int*   ei    = (const int*)d_in[1];     // edge_index [2, E]
    const float* eattr = (const float*)d_in[2];   // [E, 16]
    const float* W     = (const float*)d_in[3];   // [64, 64]
    const float* w_att = (const float*)d_in[4];   // [144]
    const float* bias  = (const float*)d_in[5];   // [64]

    const int N = in_sizes[0] / C_DIM;
    const int E = in_sizes[1] / 2;
    float* out = (float*)d_out;

    // Workspace layout (floats): h[N*64] | s1[N] | s3[N] | m[N] | ssum[N] | alpha[E]
    float* h    = (float*)d_ws;
    float* s1   = h + (size_t)N * C_DIM;
    float* s3   = s1 + N;
    float* mbuf = s3 + N;
    float* sbuf = mbuf + N;
    float* abuf = sbuf + N;

    const int TB = 256;

    // 1) out = bias broadcast; m = -inf; ssum = 0
    tml_init_kernel<<<(N * C_DIM + TB - 1) / TB, TB, 0, stream>>>(out, bias, mbuf, sbuf, N);

    // 2) h = x @ W  (fp32 WMMA, one 16x16 tile per wave, 8 waves per block)
    {
        int tilesM = (N + 15) / 16;
        int tilesN = C_DIM / 16;
        int waves  = tilesM * tilesN;
        int blocks = (waves + 7) / 8;
        tml_gemm_kernel<<<blocks, TB, 0, stream>>>(x, W, h, N);
    }

    // 3) per-node attention scalars
    tml_node_scalars_kernel<<<(N + TB - 1) / TB, TB, 0, stream>>>(h, w_att, s1, s3, N);

    // 4) alpha + segment max
    tml_alpha_kernel<<<(E + TB - 1) / TB, TB, 0, stream>>>(ei, eattr, w_att, s1, s3, abuf, mbuf, E);

    // 5) exp + segment sum
    tml_exp_kernel<<<(E + TB - 1) / TB, TB, 0, stream>>>(ei, abuf, mbuf, sbuf, E);

    // 6) normalized weighted scatter-add
    {
        long long work = (long long)E * C_DIM;
        int blocks = (int)((work + TB - 1) / TB);
        tml_scatter_kernel<<<blocks, TB, 0, stream>>>(ei, abuf, sbuf, h, out, E);
    }
}